// SceneContraction_5265629905632
// MI455X (gfx1250) — compile-verified
//
#include <hip/hip_runtime.h>
#include <math.h>

// SceneContraction: out_means = contract(means); out_cov = J cov J^T for ||x||>=1 else cov.
// Memory-bound streaming kernel: 2M pts x (48 B in + 48 B out) = 192 MB @ 23.3 TB/s => ~8.2 us floor.
// Strategy: 256-point tiles staged through LDS with CDNA5 async global<->LDS DMA
// (global_load_async_to_lds_b128 / global_store_async_from_lds_b128, ASYNCcnt),
// straight-line issue (no loops) so each wave fires its full async burst before one wait.

typedef float f4 __attribute__((ext_vector_type(4)));

#define TILE 256
#define COV_F4 (TILE * 9 / 4)   // 576 b128 transfers per tile
#define MEAN_F4 (TILE * 3 / 4)  // 192 b128 transfers per tile

__device__ __forceinline__ void async_g2l(unsigned lds_byte, const void* gptr) {
  asm volatile("global_load_async_to_lds_b128 %0, %1, off"
               :: "v"(lds_byte), "v"(gptr) : "memory");
}
__device__ __forceinline__ void async_l2g(void* gptr, unsigned lds_byte) {
  asm volatile("global_store_async_from_lds_b128 %0, %1, off"
               :: "v"(gptr), "v"(lds_byte) : "memory");
}
__device__ __forceinline__ void wait_async0() {
  asm volatile("s_wait_asynccnt 0x0" ::: "memory");
}

__device__ __forceinline__ void contract_point(float x0, float x1, float x2,
                                               const float c[9],
                                               float m[3], float o[9]) {
  const float n2 = x0 * x0 + x1 * x1 + x2 * x2;
  if (n2 < 1.0f) {
    m[0] = x0; m[1] = x1; m[2] = x2;
#pragma unroll
    for (int k = 0; k < 9; ++k) o[k] = c[k];
    return;
  }
  const float inv = 1.0f / sqrtf(n2);                    // 1/||x||
  const float a = inv * (2.0f - inv);                    // (2n-1)/n^2
  const float b = 2.0f * inv * inv * inv * (inv - 1.0f); // 2(1-n)/n^4
  m[0] = a * x0; m[1] = a * x1; m[2] = a * x2;

  // u = C x, v = C^T x, s = x^T C x
  const float u0 = c[0] * x0 + c[1] * x1 + c[2] * x2;
  const float u1 = c[3] * x0 + c[4] * x1 + c[5] * x2;
  const float u2 = c[6] * x0 + c[7] * x1 + c[8] * x2;
  const float v0 = c[0] * x0 + c[3] * x1 + c[6] * x2;
  const float v1 = c[1] * x0 + c[4] * x1 + c[7] * x2;
  const float v2 = c[2] * x0 + c[5] * x1 + c[8] * x2;
  const float s  = x0 * u0 + x1 * u1 + x2 * u2;

  const float a2 = a * a, ab = a * b, b2s = b * b * s;
  const float xs[3] = {x0, x1, x2};
  const float us[3] = {u0, u1, u2};
  const float vs[3] = {v0, v1, v2};
#pragma unroll
  for (int i = 0; i < 3; ++i) {
#pragma unroll
    for (int j = 0; j < 3; ++j) {
      // (J C J^T)_ij = a^2 C_ij + a b (u_i x_j + x_i v_j) + b^2 s x_i x_j
      o[3 * i + j] = a2 * c[3 * i + j] + ab * (us[i] * xs[j] + xs[i] * vs[j])
                   + b2s * xs[i] * xs[j];
    }
  }
}

__global__ __launch_bounds__(TILE) void scene_contraction_kernel(
    const float* __restrict__ means,   // [N,3]
    const float* __restrict__ cov,     // [N,3,3]
    float* __restrict__ out_means,     // [N,3]
    float* __restrict__ out_cov,       // [N,3,3]
    int N) {
  __shared__ __align__(16) float sM[TILE * 3];   // 3072 B
  __shared__ __align__(16) float sC[TILE * 9];   // 9216 B

  const int tid = threadIdx.x;
  const int tile0 = blockIdx.x * TILE;
  const int rem = N - tile0;
  const int count = rem < TILE ? rem : TILE;

  const float* gM = means + (size_t)tile0 * 3;
  const float* gC = cov + (size_t)tile0 * 9;
  float* oM = out_means + (size_t)tile0 * 3;
  float* oC = out_cov + (size_t)tile0 * 9;

  if (count == TILE) {
    const unsigned sc = (unsigned)(size_t)(void*)sC;  // LDS byte address (low 32 bits)
    const unsigned sm = (unsigned)(size_t)(void*)sM;

    // ---- async DMA burst: global -> LDS (16 B per lane per op, ASYNCcnt) ----
    // cov: 576 transfers = tid, tid+256, (tid<64: tid+512); means: 192 = (tid<192: tid)
    async_g2l(sc + (unsigned)tid * 16u,           (const f4*)gC + tid);
    async_g2l(sc + (unsigned)(tid + 256) * 16u,   (const f4*)gC + tid + 256);
    if (tid < COV_F4 - 512)
      async_g2l(sc + (unsigned)(tid + 512) * 16u, (const f4*)gC + tid + 512);
    if (tid < MEAN_F4)
      async_g2l(sm + (unsigned)tid * 16u,         (const f4*)gM + tid);
    wait_async0();
    __syncthreads();

    // ---- per-point compute from LDS, write back in place ----
    {
      const float x0 = sM[tid * 3 + 0];
      const float x1 = sM[tid * 3 + 1];
      const float x2 = sM[tid * 3 + 2];
      float c[9];
#pragma unroll
      for (int k = 0; k < 9; ++k) c[k] = sC[tid * 9 + k];

      float m[3], o[9];
      contract_point(x0, x1, x2, c, m, o);

      sM[tid * 3 + 0] = m[0];
      sM[tid * 3 + 1] = m[1];
      sM[tid * 3 + 2] = m[2];
#pragma unroll
      for (int k = 0; k < 9; ++k) sC[tid * 9 + k] = o[k];
    }
    __syncthreads();

    // ---- async DMA burst: LDS -> global ----
    if (tid < MEAN_F4)
      async_l2g((f4*)oM + tid,       sm + (unsigned)tid * 16u);
    async_l2g((f4*)oC + tid,         sc + (unsigned)tid * 16u);
    async_l2g((f4*)oC + tid + 256,   sc + (unsigned)(tid + 256) * 16u);
    if (tid < COV_F4 - 512)
      async_l2g((f4*)oC + tid + 512, sc + (unsigned)(tid + 512) * 16u);
    wait_async0();
  } else {
    // tail tile (only the last block): simple scalar path
    if (tid < count) {
      const float x0 = gM[tid * 3 + 0];
      const float x1 = gM[tid * 3 + 1];
      const float x2 = gM[tid * 3 + 2];
      float c[9];
#pragma unroll
      for (int k = 0; k < 9; ++k) c[k] = gC[tid * 9 + k];

      float m[3], o[9];
      contract_point(x0, x1, x2, c, m, o);

      oM[tid * 3 + 0] = m[0];
      oM[tid * 3 + 1] = m[1];
      oM[tid * 3 + 2] = m[2];
#pragma unroll
      for (int k = 0; k < 9; ++k) oC[tid * 9 + k] = o[k];
    }
  }
}

extern "C" void kernel_launch(void* const* d_in, const int* in_sizes, int n_in,
                              void* d_out, int out_size, void* d_ws, size_t ws_size,
                              hipStream_t stream) {
  const float* means = (const float*)d_in[0];   // [N,3] f32
  const float* cov   = (const float*)d_in[1];   // [N,3,3] f32
  const int N = in_sizes[0] / 3;

  float* out_means = (float*)d_out;               // first 3N floats
  float* out_cov   = out_means + (size_t)N * 3;   // next 9N floats

  const int tiles = (N + TILE - 1) / TILE;
  hipLaunchKernelGGL(scene_contraction_kernel, dim3(tiles), dim3(TILE), 0, stream,
                     means, cov, out_means, out_cov, N);
}